// SymmetricGraphPreProcessingLayer_2216203125144
// MI455X (gfx1250) — compile-verified
//
#include <hip/hip_runtime.h>
#include <stdint.h>

// Problem constants (reference: N = 8192, float32).
#define NN 8192
#define TILE 64
#define NB (NN / TILE)        // 128 tile-blocks per dimension
#define LSTR 65               // LDS row stride (floats): 65 % 64 == 1 -> conflict-free transpose

// ---- CDNA5 async global->LDS copy (gfx1250, ASYNCcnt-tracked) -----------------
// Low 32 bits of a generic pointer into LDS are the wave-relative LDS byte
// address (ISA 10.2: LDS aperture uses addr[31:0] directly).
__device__ __forceinline__ uint32_t lds_off(const void* p) {
    return (uint32_t)(uintptr_t)p;
}

__device__ __forceinline__ void async_ld_b32(uint32_t dst_lds, const float* gaddr) {
    // VDST = LDS byte address (VGPR), VADDR = 64-bit global address (VGPR pair), SADDR = off
    asm volatile("global_load_async_to_lds_b32 %0, %1, off"
                 :: "v"(dst_lds), "v"(gaddr)
                 : "memory");
}

__device__ __forceinline__ void wait_async0() {
    asm volatile("s_wait_asynccnt 0" ::: "memory");
}

// ---- Pass 1: partial row sums of a = max(adj, adj^T) with diag forced to 1 ----
// Block (bi,bj) with bi<=bj handles the tile and its mirror; contribution of
// column-block cb to row i is written to partial[cb*N + i] (each slot written
// by exactly one block -> deterministic, no atomics).
__global__ __launch_bounds__(256) void k_rowsum_partial(const float* __restrict__ adj,
                                                        float* __restrict__ partial) {
    const int bi = blockIdx.x;
    const int bj = blockIdx.y;
    if (bj < bi) return;

    __shared__ float s1[TILE * LSTR];
    __shared__ float s2[TILE * LSTR];

    const int t  = threadIdx.x;
    const int c  = t & (TILE - 1);   // column within tile (coalesced dim)
    const int r0 = t >> 6;           // 0..3

    const float* g1 = adj + (size_t)(bi * TILE) * NN + (size_t)(bj * TILE); // tile (bi,bj)
    const float* g2 = adj + (size_t)(bj * TILE) * NN + (size_t)(bi * TILE); // tile (bj,bi)

    #pragma unroll
    for (int i = 0; i < 16; ++i) {
        const int r = r0 + 4 * i;
        async_ld_b32(lds_off(&s1[r * LSTR + c]), g1 + (size_t)r * NN + c);
        async_ld_b32(lds_off(&s2[r * LSTR + c]), g2 + (size_t)r * NN + c);
    }
    wait_async0();
    __syncthreads();

    const bool diag = (bi == bj);
    #pragma unroll
    for (int i = 0; i < 16; ++i) {
        const int r = r0 + 4 * i;
        float a = s1[r * LSTR + c];
        float b = s2[c * LSTR + r];          // transpose read: banks (c+r)%64 -> conflict-free
        float m = fmaxf(a, b);
        if (diag && (r == c)) m = 1.0f;      // self-loop: diagonal forced to exactly 1
        s1[r * LSTR + c] = m;                // each (r,c) owned by exactly one thread
    }
    __syncthreads();

    if (t < TILE) {
        // row sums of m -> rows of block bi, contributed by column-block bj
        float s = 0.0f;
        #pragma unroll 8
        for (int cc = 0; cc < TILE; ++cc) s += s1[t * LSTR + cc];
        partial[(size_t)bj * NN + (size_t)(bi * TILE + t)] = s;
    } else if (t < 2 * TILE && bi != bj) {
        // column sums of m == row sums of m^T -> rows of block bj, from column-block bi
        const int cc = t - TILE;
        float s = 0.0f;
        #pragma unroll 8
        for (int rr = 0; rr < TILE; ++rr) s += s1[rr * LSTR + cc];
        partial[(size_t)bi * NN + (size_t)(bj * TILE + cc)] = s;
    }
}

// ---- Pass 2: dinv[i] = rsqrt(sum_cb partial[cb][i]) ---------------------------
__global__ __launch_bounds__(256) void k_dinv(const float* __restrict__ partial,
                                              float* __restrict__ dinv) {
    __shared__ float red[4][TILE];
    const int t    = threadIdx.x;
    const int lane = t & (TILE - 1);
    const int seg  = t >> 6;                     // 0..3, each sums 32 column-blocks
    const int i    = blockIdx.x * TILE + lane;

    float s = 0.0f;
    for (int cb = seg * 32; cb < seg * 32 + 32; ++cb)
        s += partial[(size_t)cb * NN + i];       // coalesced across lanes
    red[seg][lane] = s;
    __syncthreads();
    if (t < TILE) {
        float tot = red[0][t] + red[1][t] + red[2][t] + red[3][t];
        dinv[blockIdx.x * TILE + t] = rsqrtf(tot);
    }
}

// ---- Pass 3: recompute m per tile-pair, scale, write both orientations --------
__global__ __launch_bounds__(256) void k_scale(const float* __restrict__ adj,
                                               const float* __restrict__ dinv,
                                               float* __restrict__ out) {
    const int bi = blockIdx.x;
    const int bj = blockIdx.y;
    if (bj < bi) return;

    __shared__ float s1[TILE * LSTR];
    __shared__ float s2[TILE * LSTR];
    __shared__ float dvr[TILE];      // dinv over rows of block bi
    __shared__ float dvc[TILE];      // dinv over rows of block bj

    const int t  = threadIdx.x;
    const int c  = t & (TILE - 1);
    const int r0 = t >> 6;

    const float* g1 = adj + (size_t)(bi * TILE) * NN + (size_t)(bj * TILE);
    const float* g2 = adj + (size_t)(bj * TILE) * NN + (size_t)(bi * TILE);

    #pragma unroll
    for (int i = 0; i < 16; ++i) {
        const int r = r0 + 4 * i;
        async_ld_b32(lds_off(&s1[r * LSTR + c]), g1 + (size_t)r * NN + c);
        async_ld_b32(lds_off(&s2[r * LSTR + c]), g2 + (size_t)r * NN + c);
    }
    if (t < TILE)          dvr[t] = dinv[bi * TILE + t];
    else if (t < 2 * TILE) dvc[t - TILE] = dinv[bj * TILE + (t - TILE)];
    wait_async0();
    __syncthreads();

    const bool diag = (bi == bj);
    float mreg[16];
    #pragma unroll
    for (int i = 0; i < 16; ++i) {
        const int r = r0 + 4 * i;
        float a = s1[r * LSTR + c];
        float b = s2[c * LSTR + r];
        float m = fmaxf(a, b);
        if (diag && (r == c)) m = 1.0f;
        mreg[i] = m;
        s1[r * LSTR + c] = m;        // stage for transposed writeback
    }
    __syncthreads();

    // Tile (bi,bj): coalesced over c
    #pragma unroll
    for (int i = 0; i < 16; ++i) {
        const int r = r0 + 4 * i;
        out[(size_t)(bi * TILE + r) * NN + (size_t)(bj * TILE + c)] = mreg[i] * dvr[r] * dvc[c];
    }
    // Mirror tile (bj,bi): out[J][I] = m[c][rr] scaled; coalesced over c, LDS conflict-free
    if (!diag) {
        #pragma unroll
        for (int i = 0; i < 16; ++i) {
            const int rr = r0 + 4 * i;
            out[(size_t)(bj * TILE + rr) * NN + (size_t)(bi * TILE + c)] =
                s1[c * LSTR + rr] * dvc[rr] * dvr[c];
        }
    }
}

extern "C" void kernel_launch(void* const* d_in, const int* in_sizes, int n_in,
                              void* d_out, int out_size, void* d_ws, size_t ws_size,
                              hipStream_t stream) {
    (void)in_sizes; (void)n_in; (void)out_size; (void)ws_size;
    const float* adj = (const float*)d_in[0];
    float* out = (float*)d_out;

    // Scratch: partial row sums (NB*N floats = 4 MB) live at the start of d_out
    // (fully overwritten by pass 3); dinv (32 KB) lives in d_ws.
    float* partial = out;
    float* dinv    = (float*)d_ws;

    dim3 grid(NB, NB), block(256);
    k_rowsum_partial<<<grid, block, 0, stream>>>(adj, partial);
    k_dinv<<<NN / TILE, 256, 0, stream>>>(partial, dinv);
    k_scale<<<grid, block, 0, stream>>>(adj, dinv, out);
}